// GAT_86388972191777
// MI455X (gfx1250) — compile-verified
//
#include <hip/hip_runtime.h>
#include <hip/hip_bf16.h>
#include <math.h>

typedef __attribute__((ext_vector_type(2))) float v2f;
typedef __attribute__((ext_vector_type(8))) float v8f;

#define NEG_SLOPE 0.2f

// ---------------------------------------------------------------------------
// h = A[N,K] @ B[K,8] using V_WMMA_F32_16X16X4_F32, one wave per 16-row tile.
// B is pre-packed into LDS in the exact WMMA B lane layout so the inner loop
// is: global_load_b64 (A) + ds_load_2addr_b64 (B) + v_wmma, no EXEC churn.
//   A-layout (16x4 f32): element (m,k) -> lane = 16*(k>=2) + m, vgpr = k&1
//   B-layout (4x16 f32): element (k,n) -> lane = 16*(k>=2) + n, vgpr = k&1
//   C/D (16x16 f32): vgpr j -> row j (lanes 0-15) / j+8 (lanes 16-31), col=lane&15
// ---------------------------------------------------------------------------
template <int K>
__global__ __launch_bounds__(256)
void gat_gemm_wmma(const float* __restrict__ A, const float* __restrict__ B,
                   float* __restrict__ H, int Nrows) {
    __shared__ float sB[K * 16];               // [K/4 tiles][32 lanes][2 vgprs]
    for (int idx = (int)threadIdx.x; idx < K * 16; idx += 256) {
        int t = idx >> 6;                      // k-tile
        int l = (idx & 63) >> 1;               // lane
        int w = idx & 1;                       // vgpr within lane
        int khalf = (l >> 4) * 2;
        int n     = l & 15;
        int k     = t * 4 + khalf + w;
        sB[idx] = (n < 8) ? B[(size_t)k * 8 + n] : 0.f;
    }
    __syncthreads();

    int wave = (int)((blockIdx.x * blockDim.x + threadIdx.x) >> 5);
    int lane = (int)(threadIdx.x & 31);
    int row0 = wave * 16;
    if (row0 >= Nrows) return;                 // wave-uniform

    int mn    = lane & 15;
    int khalf = (lane >> 4) * 2;
    int arow  = row0 + mn;
    if (arow >= Nrows) arow = Nrows - 1;       // branch-free clamp (stores guarded)
    const float* Ap = A + (size_t)arow * K + khalf;
    const float* Bp = sB + lane * 2;

    v8f c = {0.f, 0.f, 0.f, 0.f, 0.f, 0.f, 0.f, 0.f};
#pragma unroll 4
    for (int k0 = 0; k0 < K; k0 += 4) {
        v2f a = *(const v2f*)(Ap + k0);
        v2f b = *(const v2f*)(Bp + (k0 << 4)); // (k0/4)*64
        c = __builtin_amdgcn_wmma_f32_16x16x4_f32(false, a, false, b,
                                                  (short)0, c, false, false);
    }

    int col   = lane & 15;
    int rbase = row0 + ((lane >> 4) ? 8 : 0);
    if (row0 + 16 <= Nrows) {
        // Fast path (always taken for N % 16 == 0): single EXEC mask (col<8),
        // then 8 unguarded clause-batched stores.
        if (col < 8) {
            float* Hp = H + (size_t)rbase * 8 + col;
#pragma unroll
            for (int j = 0; j < 8; ++j) Hp[(size_t)j * 8] = c[j];
        }
    } else if (col < 8) {
#pragma unroll
        for (int j = 0; j < 8; ++j) {
            int r = rbase + j;
            if (r < Nrows) H[(size_t)r * 8 + col] = c[j];
        }
    }
}

// ---------------------------------------------------------------------------
// Pack edge_index ([2,E] int64) + N self-loops into interleaved int32 pairs.
// ---------------------------------------------------------------------------
__global__ __launch_bounds__(256)
void gat_pack_edges(const long long* __restrict__ ei, int2* __restrict__ ed,
                    int E, int N) {
    int e = (int)(blockIdx.x * blockDim.x + threadIdx.x);
    if (e >= E + N) return;
    int2 p;
    if (e < E) { p.x = (int)ei[e]; p.y = (int)ei[(size_t)E + e]; }
    else       { p.x = p.y = e - E; }
    ed[e] = p;
}

__device__ __forceinline__ void edge_sd(const long long* __restrict__ ei,
                                        const int2* __restrict__ ed, int packed,
                                        int e, int E, int* src, int* dst) {
    if (packed) { int2 p = ed[e]; *src = p.x; *dst = p.y; }
    else if (e < E) { *src = (int)ei[e]; *dst = (int)ei[(size_t)E + e]; }
    else { *src = *dst = e - E; }
}

// ---------------------------------------------------------------------------
// Per-node prep: a_s/a_d dots, m=-inf, s=0, out initialized to bias.
// ---------------------------------------------------------------------------
__global__ __launch_bounds__(256)
void gat_attn_prep(const float* __restrict__ H,
                   const float* __restrict__ att_s, const float* __restrict__ att_d,
                   const float* __restrict__ bias,
                   float* __restrict__ a_s, float* __restrict__ a_d,
                   float* __restrict__ m, float* __restrict__ s,
                   float* __restrict__ outf, int N) {
    int i = (int)(blockIdx.x * blockDim.x + threadIdx.x);
    if (i >= N) return;
    const float4* H4 = (const float4*)(H + (size_t)i * 8);
    float4 h0 = H4[0], h1 = H4[1];
    const float4* s4 = (const float4*)att_s;
    const float4* d4 = (const float4*)att_d;
    float4 as0 = s4[0], as1 = s4[1], ad0 = d4[0], ad1 = d4[1];
    float vs = h0.x * as0.x + h0.y * as0.y + h0.z * as0.z + h0.w * as0.w
             + h1.x * as1.x + h1.y * as1.y + h1.z * as1.z + h1.w * as1.w;
    float vd = h0.x * ad0.x + h0.y * ad0.y + h0.z * ad0.z + h0.w * ad0.w
             + h1.x * ad1.x + h1.y * ad1.y + h1.z * ad1.z + h1.w * ad1.w;
    a_s[i] = vs;
    a_d[i] = vd;
    m[i] = -__builtin_inff();
    s[i] = 0.f;
    float4* o4 = (float4*)(outf + (size_t)i * 8);
    const float4* b4 = (const float4*)bias;
    o4[0] = b4[0];
    o4[1] = b4[1];
}

// Pass 1: segment max over dst of leaky(a_s[src] + a_d[dst]).
__global__ __launch_bounds__(256)
void gat_edge_max(const long long* __restrict__ ei, const int2* __restrict__ ed,
                  int packed, int E, int N,
                  const float* __restrict__ a_s, const float* __restrict__ a_d,
                  float* __restrict__ m) {
    int e = (int)(blockIdx.x * blockDim.x + threadIdx.x);
    if (e >= E + N) return;
    int src, dst;
    edge_sd(ei, ed, packed, e, E, &src, &dst);
    float v = a_s[src] + a_d[dst];
    v = (v >= 0.f) ? v : NEG_SLOPE * v;
    atomicMax(&m[dst], v);                     // global_atomic_max_num_f32
}

// Pass 2: segment sum of exp(e - m[dst]).
__global__ __launch_bounds__(256)
void gat_edge_sum(const long long* __restrict__ ei, const int2* __restrict__ ed,
                  int packed, int E, int N,
                  const float* __restrict__ a_s, const float* __restrict__ a_d,
                  const float* __restrict__ m, float* __restrict__ s) {
    int e = (int)(blockIdx.x * blockDim.x + threadIdx.x);
    if (e >= E + N) return;
    int src, dst;
    edge_sd(ei, ed, packed, e, E, &src, &dst);
    float v = a_s[src] + a_d[dst];
    v = (v >= 0.f) ? v : NEG_SLOPE * v;
    atomicAdd(&s[dst], __expf(v - m[dst]));
}

// Pass 3: out[dst] += alpha * h[src].
__global__ __launch_bounds__(256)
void gat_edge_agg(const long long* __restrict__ ei, const int2* __restrict__ ed,
                  int packed, int E, int N,
                  const float* __restrict__ a_s, const float* __restrict__ a_d,
                  const float* __restrict__ m, const float* __restrict__ s,
                  const float* __restrict__ H, float* __restrict__ outf) {
    int e = (int)(blockIdx.x * blockDim.x + threadIdx.x);
    if (e >= E + N) return;
    int src, dst;
    edge_sd(ei, ed, packed, e, E, &src, &dst);
    float v = a_s[src] + a_d[dst];
    v = (v >= 0.f) ? v : NEG_SLOPE * v;
    float alpha = __expf(v - m[dst]) / s[dst];
    const float4* H4 = (const float4*)(H + (size_t)src * 8);
    float4 h0 = H4[0], h1 = H4[1];
    float* o = outf + (size_t)dst * 8;
    atomicAdd(o + 0, alpha * h0.x);
    atomicAdd(o + 1, alpha * h0.y);
    atomicAdd(o + 2, alpha * h0.z);
    atomicAdd(o + 3, alpha * h0.w);
    atomicAdd(o + 4, alpha * h1.x);
    atomicAdd(o + 5, alpha * h1.y);
    atomicAdd(o + 6, alpha * h1.z);
    atomicAdd(o + 7, alpha * h1.w);
}

// Final 8->1 linear + LeakyReLU(0.01).
__global__ __launch_bounds__(256)
void gat_final_lin(const float* __restrict__ X, const float* __restrict__ Wl,
                   const float* __restrict__ bl, float* __restrict__ out, int N) {
    int i = (int)(blockIdx.x * blockDim.x + threadIdx.x);
    if (i >= N) return;
    const float4* X4 = (const float4*)(X + (size_t)i * 8);
    float4 x0 = X4[0], x1 = X4[1];
    const float4* W4 = (const float4*)Wl;
    float4 w0 = W4[0], w1 = W4[1];
    float acc = bl[0]
              + x0.x * w0.x + x0.y * w0.y + x0.z * w0.z + x0.w * w0.w
              + x1.x * w1.x + x1.y * w1.y + x1.z * w1.z + x1.w * w1.w;
    out[i] = (acc >= 0.f) ? acc : 0.01f * acc;
}

extern "C" void kernel_launch(void* const* d_in, const int* in_sizes, int n_in,
                              void* d_out, int out_size, void* d_ws, size_t ws_size,
                              hipStream_t stream) {
    const float*     x   = (const float*)d_in[0];      // [N, 256]
    const long long* ei  = (const long long*)d_in[1];  // [2, E] int64
    const float*     W1  = (const float*)d_in[4];      // [256, 8]
    const float*     as1 = (const float*)d_in[5];
    const float*     ad1 = (const float*)d_in[6];
    const float*     b1  = (const float*)d_in[7];
    const float*     W2  = (const float*)d_in[8];      // [8, 8]
    const float*     as2 = (const float*)d_in[9];
    const float*     ad2 = (const float*)d_in[10];
    const float*     b2  = (const float*)d_in[11];
    const float*     Wl  = (const float*)d_in[12];     // [8, 1]
    const float*     bl  = (const float*)d_in[13];

    const int N = in_sizes[0] / 256;
    const int E = in_sizes[1] / 2;

    // Workspace layout (floats): 28*N (~11.2 MB) + optional packed edges.
    float* ws  = (float*)d_ws;
    float* h   = ws;                           // [N,8] (reused by both layers)
    float* a_s = h + (size_t)N * 8;            // [N]
    float* a_d = a_s + N;                      // [N]
    float* mx  = a_d + N;                      // [N]
    float* sm  = mx + N;                       // [N]
    float* of1 = sm + N;                       // [N,8]
    float* of2 = of1 + (size_t)N * 8;          // [N,8]

    const size_t node_bytes = (size_t)N * 28 * sizeof(float);
    const size_t edge_bytes = (size_t)(E + N) * sizeof(int2);
    const int packed = (ws_size >= node_bytes + edge_bytes) ? 1 : 0;
    int2* ed = (int2*)(of2 + (size_t)N * 8);

    const int tiles = (N + 15) / 16;
    dim3 gemm_grid((tiles + 7) / 8), blk(256);
    const int eb = (E + N + 255) / 256;
    const int nb = (N + 255) / 256;

    if (packed) gat_pack_edges<<<eb, blk, 0, stream>>>(ei, ed, E, N);

    // ---- Layer 1 ----
    gat_gemm_wmma<256><<<gemm_grid, blk, 0, stream>>>(x, W1, h, N);
    gat_attn_prep<<<nb, blk, 0, stream>>>(h, as1, ad1, b1, a_s, a_d, mx, sm, of1, N);
    gat_edge_max <<<eb, blk, 0, stream>>>(ei, ed, packed, E, N, a_s, a_d, mx);
    gat_edge_sum <<<eb, blk, 0, stream>>>(ei, ed, packed, E, N, a_s, a_d, mx, sm);
    gat_edge_agg <<<eb, blk, 0, stream>>>(ei, ed, packed, E, N, a_s, a_d, mx, sm, h, of1);

    // ---- Layer 2 ----
    gat_gemm_wmma<8><<<gemm_grid, blk, 0, stream>>>(of1, W2, h, N);
    gat_attn_prep<<<nb, blk, 0, stream>>>(h, as2, ad2, b2, a_s, a_d, mx, sm, of2, N);
    gat_edge_max <<<eb, blk, 0, stream>>>(ei, ed, packed, E, N, a_s, a_d, mx);
    gat_edge_sum <<<eb, blk, 0, stream>>>(ei, ed, packed, E, N, a_s, a_d, mx, sm);
    gat_edge_agg <<<eb, blk, 0, stream>>>(ei, ed, packed, E, N, a_s, a_d, mx, sm, h, of2);

    // ---- Head ----
    gat_final_lin<<<nb, blk, 0, stream>>>(of2, Wl, bl, (float*)d_out, N);
}